// LinearChainCRF_30236569764057
// MI455X (gfx1250) — compile-verified
//
#include <hip/hip_runtime.h>
#include <hip/hip_bf16.h>
#include <math.h>
#include <stdint.h>

#define CRF_IN_DIM 768
#define CRF_NT 50
#define CRF_NC 52
#define CRF_START 50
#define CRF_END 51
#define CRF_NEG (-1000000000.0f)
#define CRF_B 64
#define CRF_L 512
#define CRF_LG_STRIDE 64   // padded logits row stride in ws
#define CRF_TT_STRIDE 53   // transposed-T LDS row stride (gcd(53,64)=1 -> conflict-free)
#define CRF_XS_STRIDE 34   // LDS x-tile row stride in dwords (32 data + 2 pad via TDM pad)

typedef __attribute__((ext_vector_type(8))) float v8f;
typedef __attribute__((ext_vector_type(2))) float v2f;
typedef __attribute__((ext_vector_type(4))) unsigned int v4u;
typedef __attribute__((ext_vector_type(8))) int v8i;
typedef __attribute__((ext_vector_type(4))) int v4i;

// ---------------------------------------------------------------------------
// Kernel 0: zero-padded bias (64 entries) so the GEMM bias add is branch-free.
// ---------------------------------------------------------------------------
__global__ __launch_bounds__(64)
void crf_setup(const float* __restrict__ bias, float* __restrict__ biasP)
{
    const int j = threadIdx.x;
    biasP[j] = (j < CRF_NT) ? bias[j] : 0.0f;
}

// ---------------------------------------------------------------------------
// TDM issue: DMA a 128-row x 32-col f32 tile of x (row stride 768) into LDS
// at byte offset lds_off, padding 2 dwords after every 32 dwords (row stride
// becomes 34 dwords -> 8B-aligned, conflict-free ds_load_b64 fragments).
// D# layout per CDNA5 ISA.
// ---------------------------------------------------------------------------
__device__ __forceinline__ void crf_tdm_load_tile(const float* gsrc, unsigned lds_off)
{
    const unsigned long long ga = (unsigned long long)(uintptr_t)gsrc;
    v4u g0;
    g0[0] = 1u;                                   // count=1, user descriptor
    g0[1] = lds_off;                              // lds_addr (bytes)
    g0[2] = (unsigned)ga;                         // global_addr[31:0]
    g0[3] = (unsigned)(ga >> 32) | (2u << 30);    // global_addr[56:32] | type=2 (image)
    v8i g1;
    g1[0] = (int)((2u << 16)      // data_size=2 -> 4 bytes
                | (1u << 20)      // pad_enable
                | (4u << 22)      // pad_interval code 4 -> every 32 dwords
                | (1u << 25));    // pad_amount  code 1 -> 2 dwords
    g1[1] = (int)(768u << 16);    // tensor_dim0[15:0]=768 in bits[79:64-ish] (bits 63:48)
    g1[2] = (int)0x80000000u;     // tensor_dim1[15:0]=32768 in bits[95:80]
    g1[3] = (int)(32u << 16);     // tile_dim0=32 in bits[127:112]
    g1[4] = 128;                  // tile_dim1=128 in bits[143:128]
    g1[5] = 768;                  // tensor_dim0_stride[31:0]=768
    g1[6] = 0;
    g1[7] = 0;
    const v4i z4 = {0, 0, 0, 0};  // 2D tensor: groups 2/3 unused
#if __clang_major__ >= 23
    const v8i z8 = {0, 0, 0, 0, 0, 0, 0, 0};
    __builtin_amdgcn_tensor_load_to_lds(g0, g1, z4, z4, z8, 0);
#else
    __builtin_amdgcn_tensor_load_to_lds(g0, g1, z4, z4, 0);
#endif
}

// ---------------------------------------------------------------------------
// Kernel 1 (TDM): lg[B*L, 64pad] = x[B*L,768] @ W[50,768]^T + bias
// f32 WMMA; x tiles DMA'd into LDS by the Tensor Data Mover, double-buffered
// (TENSORcnt), so HBM latency is fully decoupled from the WMMA loop.
// Roles: A = W (M = tag), B = x^T (N = activation row): D element v of lane l
// is logits[row][tag nt*16+8*(l>>4)+v] -> aligned v8f stores of logits rows.
// ---------------------------------------------------------------------------
__global__ __launch_bounds__(256)
void crf_gemm_logits_tdm(const float* __restrict__ x,
                         const float* __restrict__ W,
                         const float* __restrict__ biasP,
                         float* __restrict__ lg)
{
    __shared__ float xs[2][128 * CRF_XS_STRIDE];   // 2 x 17 KB double buffer

    const int tid  = threadIdx.x;
    const int wave = tid >> 5;
    const int lane = tid & 31;
    const int mrow = lane & 15;      // A row (tag within tile) / B,C column (x row)
    const int hi   = lane >> 4;      // half-wave: K pair select (K = 2*hi+{0,1})
    const int m0b  = blockIdx.x * 128;           // block's first x row
    const int lrow = wave * 16 + mrow;           // this lane's local x row

    // A operand: W rows = tags nt*16 + mrow (clamped; masked on store)
    const float* wptr[4];
    #pragma unroll
    for (int nt = 0; nt < 4; ++nt) {
        int tag = nt * 16 + mrow;
        if (tag >= CRF_NT) tag = CRF_NT - 1;
        wptr[nt] = W + (size_t)tag * CRF_IN_DIM + 2 * hi;
    }

    // preload chunk 0 (wave 0 issues; TDM is a wave-level DMA op)
    if (wave == 0)
        crf_tdm_load_tile(x + (size_t)m0b * CRF_IN_DIM,
                          (unsigned)(uintptr_t)&xs[0][0]);

    v8f acc[4] = {};
    for (int kc = 0; kc < CRF_IN_DIM; kc += 32) {
        const int buf = (kc >> 5) & 1;
        if (wave == 0) {
            __builtin_amdgcn_s_wait_tensorcnt(0);          // chunk kc landed
            if (kc + 32 < CRF_IN_DIM)                      // start chunk kc+32
                crf_tdm_load_tile(x + (size_t)m0b * CRF_IN_DIM + kc + 32,
                                  (unsigned)(uintptr_t)&xs[buf ^ 1][0]);
        }
        __syncthreads();                                   // publish chunk kc

        const float* xrow = &xs[buf][lrow * CRF_XS_STRIDE + 2 * hi];
        #pragma unroll
        for (int s = 0; s < 8; ++s) {
            const v2f bx = *(const v2f*)(xrow + 4 * s);    // B frag from LDS
            #pragma unroll
            for (int nt = 0; nt < 4; ++nt) {
                const v2f aw = *(const v2f*)(wptr[nt] + kc + 4 * s);  // A frag
                acc[nt] = __builtin_amdgcn_wmma_f32_16x16x4_f32(
                    false, aw, false, bx, (short)0, acc[nt], false, false);
            }
        }
        __syncthreads();              // all waves done with buf before reuse
    }

    // D: lane column = x row (m0b+lrow); element v = tag nt*16 + 8*hi + v
    float* orow = lg + (size_t)(m0b + lrow) * CRF_LG_STRIDE;
    #pragma unroll
    for (int nt = 0; nt < 4; ++nt) {
        const int base = nt * 16 + 8 * hi;
        #pragma unroll
        for (int v = 0; v < 8; ++v)
            acc[nt][v] += biasP[base + v];
        if (base + 7 < CRF_NT) {
            *(v8f*)(orow + base) = acc[nt];     // aligned 32B -> 2x global_store_b128
        } else {
            #pragma unroll
            for (int v = 0; v < 8; ++v)
                if (base + v < CRF_NT) orow[base + v] = acc[nt][v];
        }
    }
}

// ---------------------------------------------------------------------------
// Kernel 1 (direct-load variant, kept as reference/fallback): same math,
// x consumed straight from global with prefetch.
// ---------------------------------------------------------------------------
__global__ __launch_bounds__(256)
void crf_gemm_logits(const float* __restrict__ x,
                     const float* __restrict__ W,
                     const float* __restrict__ biasP,
                     float* __restrict__ lg)
{
    const int tid  = threadIdx.x;
    const int wave = tid >> 5;
    const int lane = tid & 31;
    const int mrow = lane & 15;
    const int hi   = lane >> 4;
    const int m0   = (blockIdx.x * 8 + wave) * 16;

    v8f acc[4] = {};
    const float* xcol = x + (size_t)(m0 + mrow) * CRF_IN_DIM + 2 * hi;
    const float* wptr[4];
    #pragma unroll
    for (int nt = 0; nt < 4; ++nt) {
        int tag = nt * 16 + mrow;
        if (tag >= CRF_NT) tag = CRF_NT - 1;
        wptr[nt] = W + (size_t)tag * CRF_IN_DIM + 2 * hi;
    }

    for (int kc = 0; kc < CRF_IN_DIM; kc += 32) {
        __builtin_prefetch(xcol + kc + 32, 0, 3);   // gfx1250 global_prefetch_b8
        #pragma unroll
        for (int s = 0; s < 8; ++s) {
            const int k4 = kc + 4 * s;
            const v2f bx = *(const v2f*)(xcol + k4);
            #pragma unroll
            for (int nt = 0; nt < 4; ++nt) {
                const v2f aw = *(const v2f*)(wptr[nt] + k4);
                acc[nt] = __builtin_amdgcn_wmma_f32_16x16x4_f32(
                    false, aw, false, bx, (short)0, acc[nt], false, false);
            }
        }
    }

    float* orow = lg + (size_t)(m0 + mrow) * CRF_LG_STRIDE;
    #pragma unroll
    for (int nt = 0; nt < 4; ++nt) {
        const int base = nt * 16 + 8 * hi;
        #pragma unroll
        for (int v = 0; v < 8; ++v)
            acc[nt][v] += biasP[base + v];
        if (base + 7 < CRF_NT) {
            *(v8f*)(orow + base) = acc[nt];
        } else {
            #pragma unroll
            for (int v = 0; v < 8; ++v)
                if (base + v < CRF_NT) orow[base + v] = acc[nt][v];
        }
    }
}

// ---------------------------------------------------------------------------
// Kernel 2: per-batch CRF forward scan + gold path score.
// ---------------------------------------------------------------------------
__global__ __launch_bounds__(64)
void crf_scan(const float* __restrict__ lg,
              const int* __restrict__ y,
              const int* __restrict__ lengths,
              const float* __restrict__ T,
              float* __restrict__ partial)
{
    __shared__ float Tt[CRF_NC * CRF_TT_STRIDE];  // Tt[j*53+i] = T[i][j]
    __shared__ float alpha[CRF_NC];
    __shared__ float red[64];

    const int bidx = blockIdx.x;
    const int j = threadIdx.x;

    for (int idx = j; idx < CRF_NC * CRF_NC; idx += 64) {
        const int i  = idx / CRF_NC;
        const int jj = idx - i * CRF_NC;
        Tt[jj * CRF_TT_STRIDE + i] = T[idx];
    }
    if (j < CRF_NC) alpha[j] = (j == CRF_START) ? 0.0f : CRF_NEG;
    __syncthreads();

    int len = lengths[bidx]; if (len < 1) len = 1;   // jnp.maximum(lengths,1)
    const float* lgb = lg + (size_t)bidx * CRF_L * CRF_LG_STRIDE;

    for (int t = 0; t < len; ++t) {
        float nv = 0.0f;
        if (j < CRF_NC) {
            const float e = (j < CRF_NT) ? lgb[t * CRF_LG_STRIDE + j] : CRF_NEG;
            const float* trow = &Tt[j * CRF_TT_STRIDE];
            float mx = -3.0e38f;
            #pragma unroll 4
            for (int i = 0; i < CRF_NC; ++i)
                mx = fmaxf(mx, alpha[i] + trow[i]);
            float s = 0.0f;
            #pragma unroll 4
            for (int i = 0; i < CRF_NC; ++i)
                s += __expf(alpha[i] + trow[i] - mx);
            nv = mx + __logf(s) + e;
        }
        __syncthreads();
        if (j < CRF_NC) alpha[j] = nv;
        __syncthreads();
    }

    const int* yb = y + bidx * CRF_L;
    float acc = 0.0f;
    for (int t = j; t < CRF_L; t += 64) {
        const float mf = (t < len) ? 1.0f : 0.0f;
        if (t < CRF_L - 1) acc += lgb[t * CRF_LG_STRIDE + yb[t]] * mf;  // emit
        if (t >= 1)        acc += T[yb[t - 1] * CRF_NC + yb[t]] * mf;   // trans
    }
    red[j] = acc;
    __syncthreads();
    for (int off = 32; off > 0; off >>= 1) {
        if (j < off) red[j] += red[j + off];
        __syncthreads();
    }

    if (j == 0) {
        float path = red[0];
        path += T[CRF_START * CRF_NC + yb[0]];
        const int   ylast  = yb[len - 1];
        const float mflast = ((CRF_L - 1) < len) ? 1.0f : 0.0f;    // reference quirk:
        path += lgb[(CRF_L - 1) * CRF_LG_STRIDE + ylast] * mflast; // emission at pos L-1
        path += T[ylast * CRF_NC + CRF_END];

        float mx = -3.0e38f;
        for (int i = 0; i < CRF_NC; ++i)
            mx = fmaxf(mx, alpha[i] + T[i * CRF_NC + CRF_END]);
        float s = 0.0f;
        for (int i = 0; i < CRF_NC; ++i)
            s += __expf(alpha[i] + T[i * CRF_NC + CRF_END] - mx);
        const float allp = mx + __logf(s);

        partial[bidx] = allp - path;
    }
}

// ---------------------------------------------------------------------------
// Kernel 3: mean over B partials -> scalar
// ---------------------------------------------------------------------------
__global__ __launch_bounds__(64)
void crf_reduce(const float* __restrict__ partial, float* __restrict__ out)
{
    __shared__ float red[64];
    const int j = threadIdx.x;
    red[j] = partial[j];
    __syncthreads();
    for (int off = 32; off > 0; off >>= 1) {
        if (j < off) red[j] += red[j + off];
        __syncthreads();
    }
    if (j == 0) out[0] = red[0] * (1.0f / (float)CRF_B);
}

extern "C" void kernel_launch(void* const* d_in, const int* in_sizes, int n_in,
                              void* d_out, int out_size, void* d_ws, size_t ws_size,
                              hipStream_t stream) {
    (void)in_sizes; (void)n_in; (void)out_size; (void)ws_size;
    const float* x       = (const float*)d_in[0];
    const int*   y       = (const int*)  d_in[1];
    const int*   lengths = (const int*)  d_in[2];
    const float* W       = (const float*)d_in[3];
    const float* bias    = (const float*)d_in[4];
    const float* T       = (const float*)d_in[5];

    float* lg      = (float*)d_ws;                                 // 32768*64 f32 = 8 MB
    float* partial = lg + (size_t)CRF_B * CRF_L * CRF_LG_STRIDE;   // 64 f32
    float* biasP   = partial + 64;                                 // 64 f32, zero-padded

    crf_setup<<<1, 64, 0, stream>>>(bias, biasP);
    // 256 blocks x 128 rows: TDM double-buffered x staging feeding f32 WMMA
    crf_gemm_logits_tdm<<<256, 256, 0, stream>>>(x, W, biasP, lg);
    crf_scan<<<CRF_B, 64, 0, stream>>>(lg, y, lengths, T, partial);
    crf_reduce<<<1, 64, 0, stream>>>(partial, (float*)d_out);
}